// TurnUtteranceVadCrossEntropyLoss_2637109919916
// MI455X (gfx1250) — compile-verified
//
#include <hip/hip_runtime.h>
#include <cstdint>

// ---------------------------------------------------------------------------
// TurnUtteranceVadCrossEntropyLoss for MI455X (gfx1250).
//
// Bandwidth-bound masked CE reduction: ~84 MB of traffic, ~1.3e8 VALU ops.
// Strategy: Tensor Data Mover (TDM) streams contiguous per-timestep slabs of
// y (14336 B) and targets (6144 B) into a double-buffered LDS ring
// (TENSORcnt-tracked, wave0 issues, barriers publish), wave32 VALU computes
// the three log-softmax NLLs from LDS, per-sample partials go to d_ws, and a
// deterministic fixed-order second kernel produces the scalar. No atomics.
// ---------------------------------------------------------------------------

typedef unsigned int v4u __attribute__((ext_vector_type(4)));
typedef int          v4i __attribute__((ext_vector_type(4)));
typedef int          v8i __attribute__((ext_vector_type(8)));

#define T_TOTAL 4096
#define B_TOTAL 512
#define NBLK    256                  // blocks in kernel 1
#define CPB     (T_TOTAL / NBLK)     // 16 timesteps per block
#define Y_SLAB  (B_TOTAL * 7 * 4)    // 14336 B, contiguous y[t,:,:]
#define T_SLAB  (B_TOTAL * 3 * 4)    // 6144 B, contiguous t[t,:,:]

// Issue a 1-D TDM copy of `nbytes` (multiple of 8, 8B-aligned both sides)
// from global memory into LDS at byte offset `lds_off`.
// D# packing per CDNA5 ISA §8.3/§8.4 (group0 128b, group1 256b):
//   g0.w0: count=1 (valid user descriptor), gather off
//   g0.w1: lds_addr (bytes, wave-relative)
//   g0.w2: global_addr[31:0]     g0.w3: global_addr[56:32] | type=2<<30
//   g1.w0: data_size=3 (8B units) at bits[17:16]; no multicast/pad/iterate
//   g1.w1[31:16]=tensor_dim0[15:0]   g1.w2[15:0]=tensor_dim0[31:16]
//   g1.w2[31:16]=tensor_dim1 lo (=1) g1.w3[31:16]=tile_dim0
//   g1.w4: tile_dim1=tile_dim2=0 (1-D tile)   g1.w5: tensor_dim0_stride lo
__device__ __forceinline__ void tdm_load_1d(unsigned lds_off,
                                            const void* gptr,
                                            unsigned nbytes) {
  const uint64_t ga    = (uint64_t)(uintptr_t)gptr;
  const unsigned units = nbytes >> 3;  // 8-byte elements
  v4u g0;
  g0.x = 1u;                                            // count=1, is_restore=0
  g0.y = lds_off;                                       // LDS byte address
  g0.z = (unsigned)ga;                                  // global addr lo32
  g0.w = (unsigned)((ga >> 32) & 0x1FFFFFFu) | (2u << 30);  // addr hi + type=2
  v8i g1;
  g1[0] = (int)(3u << 16);                              // data_size = 8 bytes
  g1[1] = (int)((units & 0xFFFFu) << 16);               // tensor_dim0[15:0]
  g1[2] = (int)(((units >> 16) & 0xFFFFu) | (1u << 16)); // dim0 hi | dim1=1
  g1[3] = (int)((units & 0xFFFFu) << 16);               // tile_dim0 = units
  g1[4] = 0;                                            // tile_dim1=tile_dim2=0
  g1[5] = (int)units;                                   // dim0 stride (unused, 1-D)
  g1[6] = 0;
  g1[7] = 0;
  v4i zz = {0, 0, 0, 0};                                // groups 2/3: <=2-D tensor
#if defined(__clang_major__) && (__clang_major__ >= 23)
  v8i z8 = {0, 0, 0, 0, 0, 0, 0, 0};
  __builtin_amdgcn_tensor_load_to_lds(g0, g1, zz, zz, z8, 0);
#else
  __builtin_amdgcn_tensor_load_to_lds(g0, g1, zz, zz, 0);
#endif
}

__device__ __forceinline__ float lse3(float a, float b, float c) {
  float m = fmaxf(a, fmaxf(b, c));
  return m + __logf(__expf(a - m) + __expf(b - m) + __expf(c - m));
}
__device__ __forceinline__ float lse2(float a, float b) {
  float m = fmaxf(a, b);
  return m + __logf(__expf(a - m) + __expf(b - m));
}

// NLL of the three heads (widths 3, 2, 2) for one (t, b) element.
__device__ __forceinline__ float elem_nll(const float* __restrict__ y7,
                                          const int* __restrict__ t3) {
  float l0 = y7[0], l1 = y7[1], l2 = y7[2];
  float l3 = y7[3], l4 = y7[4], l5 = y7[5], l6 = y7[6];
  int a = t3[0], b = t3[1], c = t3[2];
  float pa = (a == 0) ? l0 : ((a == 1) ? l1 : l2);
  float n0 = lse3(l0, l1, l2) - pa;
  float n1 = lse2(l3, l4) - ((b == 0) ? l3 : l4);
  float n2 = lse2(l5, l6) - ((c == 0) ? l5 : l6);
  return n0 + n1 + n2;
}

// Kernel 1: per-block partial per-sample masked-CE sums over CPB timesteps.
// TDM double-buffers y/t slabs through LDS; 40 KB static LDS.
// NOTE: all LDS buffer addresses are derived arithmetically from the single
// shared-array base at runtime (no pointer tables -> no addrspacecast in a
// static initializer, which ld.lld rejects).
__global__ void __launch_bounds__(256)
k_partials(const float* __restrict__ y, const int* __restrict__ tg,
           const int* __restrict__ ylen, float* __restrict__ ws) {
  __shared__ __align__(16) unsigned char smem[2 * (Y_SLAB + T_SLAB)];  // 40960 B

  unsigned char* smem_p   = smem;                         // generic pointer (runtime cast)
  const unsigned lds_base = (unsigned)(uintptr_t)smem_p;  // wave-relative LDS byte addr

  const int  tid    = threadIdx.x;
  const int  t0     = blockIdx.x * CPB;
  const bool issuer = (tid < 32);  // wave 0 owns the TDM queue (per-wave TENSORcnt)

  const int len0 = ylen[tid];
  const int len1 = ylen[tid + 256];
  float acc0 = 0.0f, acc1 = 0.0f;

  if (issuer) {
    tdm_load_1d(lds_base,              y  + (size_t)t0 * B_TOTAL * 7, Y_SLAB);
    tdm_load_1d(lds_base + 2 * Y_SLAB, tg + (size_t)t0 * B_TOTAL * 3, T_SLAB);
  }

  for (int i = 0; i < CPB; ++i) {
    const unsigned cur  = (unsigned)(i & 1);
    const unsigned yoff = cur * Y_SLAB;                   // 0 or Y_SLAB
    const unsigned toff = 2u * Y_SLAB + cur * T_SLAB;     // target ring base
    if (issuer) {
      if (i + 1 < CPB) {
        // Prefetch next slab into the other buffer, then wait only for the
        // current slab's two DMAs (TDM ops complete in order per wave).
        const unsigned nyoff = (cur ^ 1u) * Y_SLAB;
        const unsigned ntoff = 2u * Y_SLAB + (cur ^ 1u) * T_SLAB;
        tdm_load_1d(lds_base + nyoff, y  + (size_t)(t0 + i + 1) * B_TOTAL * 7, Y_SLAB);
        tdm_load_1d(lds_base + ntoff, tg + (size_t)(t0 + i + 1) * B_TOTAL * 3, T_SLAB);
        __builtin_amdgcn_s_wait_tensorcnt(2);
      } else {
        __builtin_amdgcn_s_wait_tensorcnt(0);
      }
    }
    __syncthreads();  // slab i is now visible to all 8 waves

    const int    t_global = t0 + i;
    const float* yb = (const float*)(smem_p + yoff);
    const int*   tb = (const int*)(smem_p + toff);
    // strides 7 / 3 dwords are coprime with 64 banks -> conflict-free
    float n0 = elem_nll(yb + (size_t)tid * 7,         tb + (size_t)tid * 3);
    float n1 = elem_nll(yb + (size_t)(tid + 256) * 7, tb + (size_t)(tid + 256) * 3);
    if (t_global < len0) acc0 += n0;
    if (t_global < len1) acc1 += n1;

    __syncthreads();  // all reads of buf[cur] done before it is refilled
  }

  ws[(size_t)blockIdx.x * B_TOTAL + tid]       = acc0;
  ws[(size_t)blockIdx.x * B_TOTAL + tid + 256] = acc1;
}

// Kernel 2: deterministic fixed-order final reduction -> scalar loss.
__global__ void __launch_bounds__(512)
k_reduce(const int* __restrict__ ylen, const float* __restrict__ ws,
         float* __restrict__ out) {
  __shared__ float red[B_TOTAL];
  const int b = threadIdx.x;
  float s = 0.0f;
  for (int k = 0; k < NBLK; ++k) s += ws[(size_t)k * B_TOTAL + b];
  const int len = ylen[b];
  red[b] = (len > 0) ? (s / (float)len) : 0.0f;
  __syncthreads();
  for (int off = 256; off > 0; off >>= 1) {
    if (b < off) red[b] += red[b + off];
    __syncthreads();
  }
  if (b == 0) out[0] = red[0] / (float)B_TOTAL;
}

extern "C" void kernel_launch(void* const* d_in, const int* in_sizes, int n_in,
                              void* d_out, int out_size, void* d_ws, size_t ws_size,
                              hipStream_t stream) {
  const float* y    = (const float*)d_in[0];
  const int*   t    = (const int*)d_in[1];
  const int*   ylen = (const int*)d_in[2];
  float*       ws   = (float*)d_ws;   // needs NBLK*B_TOTAL*4 = 512 KB
  (void)in_sizes; (void)n_in; (void)out_size; (void)ws_size;

  k_partials<<<NBLK, 256, 0, stream>>>(y, t, ylen, ws);
  k_reduce<<<1, 512, 0, stream>>>(ylen, ws, (float*)d_out);
}